// CLIFFHead_36447092474477
// MI455X (gfx1250) — compile-verified
//
#include <hip/hip_runtime.h>
#include <hip/hip_bf16.h>
#include <math.h>

// ---------------- problem constants ----------------
#define Bc   4096
#define DZc  2048
#define DEc  128
#define Mc   32
#define Hc   1024
#define W1Kc (DZc + DEc)   // 2176, row stride of W1
#define KC   16            // K-chunk staged in LDS (double buffered)
#define KP   20            // padded LDS row stride (floats): 20*l mod 64 -> conflict-free

typedef float v2f __attribute__((ext_vector_type(2)));
typedef float v8f __attribute__((ext_vector_type(8)));

static __device__ __forceinline__ float gelu_exact(float x) {
    return 0.5f * x * (1.0f + erff(x * 0.70710678118654752440f));
}

// Async global->LDS copy, 16B per lane. Tracked by ASYNCcnt (gfx1250).
// LDS address = low 32 bits of the generic pointer.
static __device__ __forceinline__ void async_copy_b128(const float* gptr, float* lptr) {
    const unsigned lds = (unsigned)(unsigned long long)(uintptr_t)lptr;
    asm volatile("global_load_async_to_lds_b128 %0, %1, off"
                 :: "v"(lds), "v"(gptr)
                 : "memory");
}
static __device__ __forceinline__ void wait_async0() {
    asm volatile("s_wait_asynccnt 0x0" ::: "memory");
}

// ---------------- En = l2norm(E) rows: 32 blocks x 32 threads ----------------
__global__ void cliff_en(const float* __restrict__ E, float* __restrict__ En) {
    const int m = blockIdx.x;
    const int lane = threadIdx.x;          // 0..31
    float v[4];
    float s = 0.f;
    #pragma unroll
    for (int j = 0; j < 4; ++j) {
        v[j] = E[m * DEc + j * 32 + lane];
        s += v[j] * v[j];
    }
    #pragma unroll
    for (int d = 1; d < 32; d <<= 1) s += __shfl_xor(s, d, 32);
    const float inv = 1.0f / fmaxf(sqrtf(s), 1e-12f);
    #pragma unroll
    for (int j = 0; j < 4; ++j) En[m * DEc + j * 32 + lane] = v[j] * inv;
}

// ---------------- c1[m,h] = sum_e E[m,e]*W1[m,h,DZ+e] + b1[m,h] ----------------
__global__ __launch_bounds__(256) void cliff_c1(const float* __restrict__ E,
                                                const float* __restrict__ W1,
                                                const float* __restrict__ b1,
                                                float* __restrict__ c1) {
    __shared__ float Es[DEc];
    const int m = blockIdx.y;
    const int h = blockIdx.x * 256 + threadIdx.x;
    if (threadIdx.x < DEc) Es[threadIdx.x] = E[m * DEc + threadIdx.x];
    __syncthreads();
    const float* wrow = W1 + ((size_t)m * Hc + h) * W1Kc + DZc;
    float s = 0.f;
    #pragma unroll 8
    for (int e = 0; e < DEc; ++e) s += Es[e] * wrow[e];
    c1[m * Hc + h] = s + b1[m * Hc + h];
}

// ---------------- bbase[r,o] = z[r]·base_W[o] + base_b[o] : 1 wave / row ----------------
__global__ __launch_bounds__(256) void cliff_base(const float* __restrict__ z,
                                                  const float* __restrict__ base_W,
                                                  const float* __restrict__ base_b,
                                                  float* __restrict__ bbase) {
    const int lane = threadIdx.x & 31;
    const int wv   = threadIdx.x >> 5;
    const int r    = blockIdx.x * 8 + wv;
    const float* zr = z + (size_t)r * DZc;
    float s0 = 0.f, s1 = 0.f, s2 = 0.f;
    for (int k = lane; k < DZc; k += 32) {
        const float zv = zr[k];
        s0 += zv * base_W[k];
        s1 += zv * base_W[DZc + k];
        s2 += zv * base_W[2 * DZc + k];
    }
    #pragma unroll
    for (int d = 1; d < 32; d <<= 1) {
        s0 += __shfl_xor(s0, d, 32);
        s1 += __shfl_xor(s1, d, 32);
        s2 += __shfl_xor(s2, d, 32);
    }
    if (lane == 0) {
        bbase[r * 3 + 0] = s0 + base_b[0];
        bbase[r * 3 + 1] = s1 + base_b[1];
        bbase[r * 3 + 2] = s2 + base_b[2];
    }
}

// ---------------- gate GEMM (WMMA f32) + l2norm + sims, fused ----------------
// block: 16 rows x all 128 gate cols, 8 waves each own a 16x16 tile.
// Async double-buffered LDS staging (same ASYNCcnt pipeline as cliff_main).
__global__ __launch_bounds__(256) void cliff_gate_sims(const float* __restrict__ z,
                                                       const float* __restrict__ gate_W,
                                                       const float* __restrict__ gate_b,
                                                       const float* __restrict__ En,
                                                       float* __restrict__ sims) {
    __shared__ float zs[2][16 * KP];
    __shared__ float gw[2][DEc * KP];
    __shared__ float qs[16 * DEc];
    __shared__ float Ens[Mc * DEc];
    __shared__ float rnorm[16];

    const int tid  = threadIdx.x;
    const int lane = tid & 31;
    const int wv   = tid >> 5;
    const int r0   = blockIdx.x * 16;

    auto issue = [&](int kk, int b) {
        if (tid < 64) {                      // zs: 16 rows x 4 float4
            const int row = tid >> 2, c4 = tid & 3;
            async_copy_b128(z + (size_t)(r0 + row) * DZc + kk + c4 * 4,
                            &zs[b][row * KP + c4 * 4]);
        }
        #pragma unroll
        for (int j = 0; j < 2; ++j) {        // gw: 128 rows x 4 float4
            const int i = tid + j * 256;
            const int row = i >> 2, c4 = i & 3;
            async_copy_b128(gate_W + (size_t)row * DZc + kk + c4 * 4,
                            &gw[b][row * KP + c4 * 4]);
        }
    };

    for (int i = tid; i < Mc * DEc; i += 256) Ens[i] = En[i];

    v8f acc = {0.f, 0.f, 0.f, 0.f, 0.f, 0.f, 0.f, 0.f};

    issue(0, 0);
    int buf = 0;
    for (int kk = 0; kk < DZc; kk += KC, buf ^= 1) {
        wait_async0();        // my chunk-kk copies have landed
        __syncthreads();      // everyone's data visible; buf^1 compute done
        if (kk + KC < DZc) issue(kk + KC, buf ^ 1);

        #pragma unroll
        for (int k = 0; k < KC; k += 4) {
            const int koff = k + ((lane >> 4) << 1);
            const v2f a = *(const v2f*)&zs[buf][(lane & 15) * KP + koff];
            const v2f b = *(const v2f*)&gw[buf][(wv * 16 + (lane & 15)) * KP + koff];
            acc = __builtin_amdgcn_wmma_f32_16x16x4_f32(false, a, false, b,
                                                        (short)0, acc, false, false);
        }
    }
    // write q tile (+ gate_b) to LDS
    const int col = wv * 16 + (lane & 15);
    const float gb = gate_b[col];
    #pragma unroll
    for (int v = 0; v < 8; ++v) {
        const int row = v + ((lane >> 4) << 3);
        qs[row * DEc + col] = acc[v] + gb;
    }
    __syncthreads();
    if (tid < 16) {
        float s = 0.f;
        for (int c = 0; c < DEc; ++c) { const float q = qs[tid * DEc + c]; s += q * q; }
        rnorm[tid] = 1.0f / fmaxf(sqrtf(s), 1e-12f);
    }
    __syncthreads();
    const int r  = tid >> 4;
    const int mb = tid & 15;
    #pragma unroll
    for (int half = 0; half < 2; ++half) {
        const int mi = mb + half * 16;
        float s = 0.f;
        #pragma unroll 8
        for (int c = 0; c < DEc; ++c) s += qs[r * DEc + c] * Ens[mi * DEc + c];
        sims[(size_t)(r0 + r) * Mc + mi] = s * rnorm[r];
    }
}

// ---------------- main fused GEMM: h = gelu(z@W1z^T + c1), delta += h@W2^T ----------------
// grid = (B/128, H/256, M). block = 256 thr = 8 waves; wave w owns cols [w*32, w*32+32).
// Double-buffered LDS fed by GLOBAL_LOAD_ASYNC_TO_LDS_B128 (ASYNCcnt pipeline).
__global__ __launch_bounds__(256) void cliff_main(const float* __restrict__ z,
                                                  const float* __restrict__ W1,
                                                  const float* __restrict__ W2,
                                                  const float* __restrict__ c1,
                                                  float* __restrict__ outL) {
    __shared__ float zs [2][128 * KP];
    __shared__ float ws1[2][256 * KP];
    __shared__ float dacc[128 * 3];

    const int tid   = threadIdx.x;
    const int lane  = tid & 31;
    const int wv    = tid >> 5;
    const int r0    = blockIdx.x * 128;
    const int hbase = blockIdx.y * 256;
    const int m     = blockIdx.z;

    const float* zbase = z + (size_t)r0 * DZc;
    const float* wbase = W1 + ((size_t)m * Hc + hbase) * W1Kc;

    // issue async copies for one K-chunk into LDS buffer b (6 x b128 per thread)
    auto issue = [&](int kk, int b) {
        #pragma unroll
        for (int j = 0; j < 2; ++j) {            // zs: 128 rows x 4 float4
            const int i = tid + j * 256;
            const int row = i >> 2, c4 = i & 3;
            async_copy_b128(zbase + (size_t)row * DZc + kk + c4 * 4,
                            &zs[b][row * KP + c4 * 4]);
        }
        #pragma unroll
        for (int j = 0; j < 4; ++j) {            // ws1: 256 rows x 4 float4
            const int i = tid + j * 256;
            const int row = i >> 2, c4 = i & 3;
            async_copy_b128(wbase + (size_t)row * W1Kc + kk + c4 * 4,
                            &ws1[b][row * KP + c4 * 4]);
        }
    };

    for (int i = tid; i < 128 * 3; i += 256) dacc[i] = 0.f;

    v8f acc[8][2];
    #pragma unroll
    for (int rt = 0; rt < 8; ++rt)
        #pragma unroll
        for (int ct = 0; ct < 2; ++ct)
            acc[rt][ct] = (v8f){0.f, 0.f, 0.f, 0.f, 0.f, 0.f, 0.f, 0.f};

    issue(0, 0);                                  // prologue prefetch

    int buf = 0;
    for (int kk = 0; kk < DZc; kk += KC, buf ^= 1) {
        wait_async0();        // my chunk-kk copies have landed in LDS
        __syncthreads();      // all threads' chunk-kk data visible; buf^1 compute done
        if (kk + KC < DZc) issue(kk + KC, buf ^ 1);   // overlap with compute below

        #pragma unroll
        for (int k = 0; k < KC; k += 4) {
            const int koff = k + ((lane >> 4) << 1);
            v2f a[8], bb[2];
            #pragma unroll
            for (int rt = 0; rt < 8; ++rt)
                a[rt] = *(const v2f*)&zs[buf][(rt * 16 + (lane & 15)) * KP + koff];
            #pragma unroll
            for (int ct = 0; ct < 2; ++ct)
                bb[ct] = *(const v2f*)&ws1[buf][(wv * 32 + ct * 16 + (lane & 15)) * KP + koff];
            #pragma unroll
            for (int rt = 0; rt < 8; ++rt)
                #pragma unroll
                for (int ct = 0; ct < 2; ++ct)
                    acc[rt][ct] = __builtin_amdgcn_wmma_f32_16x16x4_f32(
                        false, a[rt], false, bb[ct], (short)0, acc[rt][ct], false, false);
        }
    }
    __syncthreads();

    // epilogue: + c1, gelu, × W2 columns, reduce over this wave's 32 h-cols
    const int colB = hbase + wv * 32 + (lane & 15);
    const float c1v0 = c1[m * Hc + colB];
    const float c1v1 = c1[m * Hc + colB + 16];
    float w2v[3][2];
    #pragma unroll
    for (int o = 0; o < 3; ++o) {
        w2v[o][0] = W2[((size_t)m * 3 + o) * Hc + colB];
        w2v[o][1] = W2[((size_t)m * 3 + o) * Hc + colB + 16];
    }
    #pragma unroll
    for (int rt = 0; rt < 8; ++rt) {
        #pragma unroll
        for (int v = 0; v < 8; ++v) {
            const float h0 = gelu_exact(acc[rt][0][v] + c1v0);
            const float h1 = gelu_exact(acc[rt][1][v] + c1v1);
            const int row = rt * 16 + v + ((lane >> 4) << 3);
            #pragma unroll
            for (int o = 0; o < 3; ++o) {
                float p = h0 * w2v[o][0] + h1 * w2v[o][1];
                p += __shfl_xor(p, 1, 16);
                p += __shfl_xor(p, 2, 16);
                p += __shfl_xor(p, 4, 16);
                p += __shfl_xor(p, 8, 16);
                if ((lane & 15) == 0) atomicAdd(&dacc[row * 3 + o], p);
            }
        }
    }
    __syncthreads();
    for (int i = tid; i < 128 * 3; i += 256) {
        const int row = i / 3, o = i - row * 3;
        atomicAdd(&outL[(size_t)(r0 + row) * (Mc * 3) + m * 3 + o], dacc[i]);
    }
}

// ---------------- L[i] += bbase + b2 ----------------
__global__ __launch_bounds__(256) void cliff_final(const float* __restrict__ bbase,
                                                   const float* __restrict__ b2,
                                                   float* __restrict__ outL) {
    const int i = blockIdx.x * 256 + threadIdx.x;   // over B*96
    const int r = i / 96;
    const int c = i - r * 96;
    outL[i] += bbase[r * 3 + (c % 3)] + b2[c];
}

// ---------------- launcher ----------------
extern "C" void kernel_launch(void* const* d_in, const int* in_sizes, int n_in,
                              void* d_out, int out_size, void* d_ws, size_t ws_size,
                              hipStream_t stream) {
    (void)in_sizes; (void)n_in; (void)out_size; (void)ws_size;
    const float* z      = (const float*)d_in[0];
    const float* base_W = (const float*)d_in[1];
    const float* base_b = (const float*)d_in[2];
    const float* gate_W = (const float*)d_in[3];
    const float* gate_b = (const float*)d_in[4];
    const float* E      = (const float*)d_in[5];
    const float* W1     = (const float*)d_in[6];
    const float* b1     = (const float*)d_in[7];
    const float* W2     = (const float*)d_in[8];
    const float* b2     = (const float*)d_in[9];

    float* outF = (float*)d_out;
    float* L    = outF;                          // B x 96
    float* sims = outF + (size_t)Bc * (Mc * 3);  // B x 32

    float* En    = (float*)d_ws;                 // 32*128
    float* c1    = En + Mc * DEc;                // 32*1024
    float* bbase = c1 + Mc * Hc;                 // 4096*3

    hipMemsetAsync(L, 0, (size_t)Bc * (Mc * 3) * sizeof(float), stream);

    cliff_en   <<<Mc, 32, 0, stream>>>(E, En);
    cliff_c1   <<<dim3(Hc / 256, Mc), 256, 0, stream>>>(E, W1, b1, c1);
    cliff_base <<<Bc / 8, 256, 0, stream>>>(z, base_W, base_b, bbase);
    cliff_gate_sims<<<Bc / 16, 256, 0, stream>>>(z, gate_W, gate_b, En, sims);
    cliff_main <<<dim3(Bc / 128, Hc / 256, Mc), 256, 0, stream>>>(z, W1, W2, c1, L);
    cliff_final<<<(Bc * (Mc * 3)) / 256, 256, 0, stream>>>(bbase, b2, L);
}